// GraphSAGE_86397562126633
// MI455X (gfx1250) — compile-verified
//
#include <hip/hip_runtime.h>

// ---------------------------------------------------------------------------
// GraphSAGE 2-layer forward for MI455X (gfx1250, wave32).
//   N=40000 nodes, E=640000 edges, IN=128, HID=128, OUT=64.
// Aggregation = L2-resident f32 atomic scatter (dominant cost).
// GEMMs use V_WMMA_F32_16X16X4_F32 to keep full f32 reference precision
// (GEMM FLOPs ~4 GF total -> far from the matrix-core roof either way).
// ---------------------------------------------------------------------------

typedef __attribute__((ext_vector_type(2))) float v2f;
typedef __attribute__((ext_vector_type(8))) float v8f;

#define N_NODES 40000
#define E_EDGES 640000
#define IN_F    128
#define HID     128
#define OUT_F   64

// ---------------------------------------------------------------------------
__global__ __launch_bounds__(256)
void zero_f32(float* __restrict__ p, long n) {
  long i = (long)blockIdx.x * blockDim.x + threadIdx.x;
  if (i < n) p[i] = 0.0f;
}

// ---------------------------------------------------------------------------
// One wave (32 lanes) per edge: each lane moves one float4 (4 of 128 feats).
// Gathers x[src] (L2-resident) and atomically accumulates into agg[dst].
// Lane 0 bumps the in-degree counter.
__global__ __launch_bounds__(256)
void scatter_mean_accum(const float* __restrict__ x,
                        const long long* __restrict__ ei,  // [2, E] int64
                        float* __restrict__ agg,
                        float* __restrict__ cnt,
                        int nEdges) {
  int wave = (int)((blockIdx.x * (long)blockDim.x + threadIdx.x) >> 5);
  int lane = threadIdx.x & 31;
  if (wave >= nEdges) return;
  long long s = ei[wave];
  long long d = ei[(long)nEdges + wave];
  const float4* xs = (const float4*)(x + s * (long)IN_F);
  float*        ad = agg + d * (long)IN_F;
  float4 v = xs[lane];                 // 32 lanes * float4 = 128 feats
  atomicAdd(ad + lane * 4 + 0, v.x);
  atomicAdd(ad + lane * 4 + 1, v.y);
  atomicAdd(ad + lane * 4 + 2, v.z);
  atomicAdd(ad + lane * 4 + 3, v.w);
  if (lane == 0) atomicAdd(cnt + d, 1.0f);
}

// ---------------------------------------------------------------------------
// Fused SAGEConv GEMM: out = act( (agg/max(cnt,1)) @ Wl + bias + xin @ Wr )
// One wave computes one 16x16 output tile via V_WMMA_F32_16X16X4_F32.
//
// Operand layouts (ISA 7.12.2, 32-bit, wave32):
//   A (16x4, MxK):  lane L -> M = L%16 ; VGPR0/1 hold K = 2*(L/16), 2*(L/16)+1
//   B (4x16, KxN):  lane L -> N = L%16 ; VGPR0/1 hold K = 2*(L/16), 2*(L/16)+1
//   C/D (16x16):    VGPR r -> M = r + 8*(L/16), N = L%16
template <int K, int NOUT, bool RELU>
__global__ __launch_bounds__(256)
void sage_gemm_wmma(const float* __restrict__ agg,
                    const float* __restrict__ cnt,
                    const float* __restrict__ xin,
                    const float* __restrict__ Wl,
                    const float* __restrict__ Wr,
                    const float* __restrict__ bias,
                    float* __restrict__ out,
                    int nRows) {
  const int waveInBlk = threadIdx.x >> 5;
  const int lane      = threadIdx.x & 31;
  const int tilesN    = NOUT >> 4;
  const int tile      = blockIdx.x * 8 + waveInBlk;   // 8 waves / block
  const int mTile     = tile / tilesN;
  const int nTile     = tile % tilesN;
  if (mTile * 16 >= nRows) return;                    // uniform per wave

  const int m  = mTile * 16 + (lane & 15);   // A row for this lane
  const int kg = lane >> 4;                  // K sub-group: 0 or 1
  const int n  = nTile * 16 + (lane & 15);   // B/C column for this lane

  const float invc = 1.0f / fmaxf(cnt[m], 1.0f);
  const float* aggRow = agg + (long)m * K;
  const float* xRow   = xin + (long)m * K;

  v8f acc = {};
#pragma unroll
  for (int k = 0; k < K; k += 4) {
    const int ka = k + 2 * kg;
    // --- term 1: mean-aggregated neighbors @ Wl ---
    v2f a1; a1.x = aggRow[ka] * invc;      a1.y = aggRow[ka + 1] * invc;
    v2f b1; b1.x = Wl[ka * NOUT + n];      b1.y = Wl[(ka + 1) * NOUT + n];
    acc = __builtin_amdgcn_wmma_f32_16x16x4_f32(false, a1, false, b1,
                                                (short)0, acc, false, false);
    // --- term 2: root features @ Wr ---
    v2f a2; a2.x = xRow[ka];               a2.y = xRow[ka + 1];
    v2f b2; b2.x = Wr[ka * NOUT + n];      b2.y = Wr[(ka + 1) * NOUT + n];
    acc = __builtin_amdgcn_wmma_f32_16x16x4_f32(false, a2, false, b2,
                                                (short)0, acc, false, false);
  }

  const float bv = bias[n];
#pragma unroll
  for (int r = 0; r < 8; ++r) {
    float v = acc[r] + bv;
    if (RELU) v = fmaxf(v, 0.0f);
    const int row = mTile * 16 + r + 8 * kg;          // C/D layout
    out[(long)row * NOUT + n] = v;
  }
}

// ---------------------------------------------------------------------------
extern "C" void kernel_launch(void* const* d_in, const int* in_sizes, int n_in,
                              void* d_out, int out_size, void* d_ws, size_t ws_size,
                              hipStream_t stream) {
  // setup_inputs order: x, edge_index(int64), W1l, b1, W1r, W2l, b2, W2r
  const float*     x   = (const float*)d_in[0];
  const long long* ei  = (const long long*)d_in[1];
  const float*     W1l = (const float*)d_in[2];
  const float*     b1  = (const float*)d_in[3];
  const float*     W1r = (const float*)d_in[4];
  const float*     W2l = (const float*)d_in[5];
  const float*     b2  = (const float*)d_in[6];
  const float*     W2r = (const float*)d_in[7];
  float*           out = (float*)d_out;

  // Workspace layout: agg[N*128] | cnt[N] | h[N*128]   (~41 MB)
  float* agg = (float*)d_ws;
  float* cnt = agg + (size_t)N_NODES * HID;
  float* h   = cnt + N_NODES;

  const long nZero = (long)N_NODES * HID + N_NODES;   // agg + cnt contiguous
  const int  zB    = (int)((nZero + 255) / 256);
  const int  sB    = (E_EDGES * 32 + 255) / 256;      // 1 wave per edge

  // ---- Layer 1: h = relu(mean_agg(x) @ W1l + b1 + x @ W1r) ----
  zero_f32<<<zB, 256, 0, stream>>>(agg, nZero);
  scatter_mean_accum<<<sB, 256, 0, stream>>>(x, ei, agg, cnt, E_EDGES);
  {
    const int tiles = (N_NODES / 16) * (HID / 16);    // 20000, divisible by 8
    sage_gemm_wmma<IN_F, HID, true><<<tiles / 8, 256, 0, stream>>>(
        agg, cnt, x, W1l, W1r, b1, h, N_NODES);
  }

  // ---- Layer 2: out = mean_agg(h) @ W2l + b2 + h @ W2r ----
  zero_f32<<<zB, 256, 0, stream>>>(agg, nZero);
  scatter_mean_accum<<<sB, 256, 0, stream>>>(h, ei, agg, cnt, E_EDGES);
  {
    const int tiles = (N_NODES / 16) * (OUT_F / 16);  // 10000, divisible by 8
    sage_gemm_wmma<HID, OUT_F, false><<<tiles / 8, 256, 0, stream>>>(
        agg, cnt, h, W2l, W2r, b2, out, N_NODES);
  }
}